// GCN_35141422415926
// MI455X (gfx1250) — compile-verified
//
#include <hip/hip_runtime.h>

#define N_NODES 100000
#define IN_C    128
#define HID     64
#define OUT_C   2
#define N_EDGES 1600000

typedef float v2f __attribute__((ext_vector_type(2)));
typedef float v8f __attribute__((ext_vector_type(8)));

// ---------------------------------------------------------------------------
// Degree: deg[col[e]] += 1  (self loops folded in later as +1)
// ---------------------------------------------------------------------------
__global__ void gcn_deg_kernel(const long long* __restrict__ col,
                               float* __restrict__ deg) {
    int e = blockIdx.x * blockDim.x + threadIdx.x;
    if (e < N_EDGES) atomicAdd(&deg[(int)col[e]], 1.0f);
}

// dinv[i] = rsqrt(deg[i] + 1)   (in place; +1 = self loop, so always > 0)
__global__ void gcn_dinv_kernel(float* __restrict__ deg) {
    int i = blockIdx.x * blockDim.x + threadIdx.x;
    if (i < N_NODES) deg[i] = rsqrtf(deg[i] + 1.0f);
}

// norm[e] = dinv[row[e]] * dinv[col[e]]  (shared by both layers)
__global__ void gcn_norm_kernel(const long long* __restrict__ ei,
                                const float* __restrict__ dinv,
                                float* __restrict__ norm) {
    int e = blockIdx.x * blockDim.x + threadIdx.x;
    if (e < N_EDGES)
        norm[e] = dinv[(int)ei[e]] * dinv[(int)ei[N_EDGES + e]];
}

// ---------------------------------------------------------------------------
// GEMM1: h1 = x @ W1  ([100000,128] @ [128,64]) via V_WMMA_F32_16X16X4_F32.
// 256 threads = 8 waves; each wave owns one 16x16 tile (2 M-tiles x 4 N-tiles
// => block covers 32 rows x 64 cols). W1 + x-tile staged in LDS.
// ---------------------------------------------------------------------------
__global__ void __launch_bounds__(256)
gcn_gemm1_wmma(const float* __restrict__ x,
               const float* __restrict__ W1,
               float* __restrict__ h1) {
    __shared__ float sW[IN_C * HID];   // 32 KB
    __shared__ float sX[32 * IN_C];    // 16 KB
    const int tid = threadIdx.x;
    const int xbase = blockIdx.x * 32 * IN_C;
    for (int i = tid; i < IN_C * HID; i += 256) sW[i] = W1[i];
    for (int i = tid; i < 32 * IN_C;  i += 256) sX[i] = x[xbase + i];
    __syncthreads();

    const int wave   = tid >> 5;       // 0..7
    const int lane   = tid & 31;
    const int nTile  = wave & 3;       // cols 0/16/32/48
    const int mLocal = wave >> 2;      // rows 0/16 within block
    const int laneLo = lane & 15;
    const int laneHi = lane >> 4;      // K-half select for A/B
    const int aRow   = mLocal * 16 + laneLo;   // A: M = lane%16
    const int col0   = nTile * 16;

    v8f c = {};
#pragma unroll
    for (int k0 = 0; k0 < IN_C; k0 += 4) {
        const int k = k0 + 2 * laneHi;
        v2f a, b;
        a.x = sX[aRow * IN_C + k];
        a.y = sX[aRow * IN_C + k + 1];
        b.x = sW[(k    ) * HID + col0 + laneLo];
        b.y = sW[(k + 1) * HID + col0 + laneLo];
        // 8 args: (neg_a, A, neg_b, B, c_mod, C, reuse_a, reuse_b)
        c = __builtin_amdgcn_wmma_f32_16x16x4_f32(false, a, false, b,
                                                  (short)0, c, false, false);
    }

    // C layout: VGPR r holds M = r + 8*laneHi, N = laneLo
    const int gRow0 = blockIdx.x * 32 + mLocal * 16;
#pragma unroll
    for (int r = 0; r < 8; ++r)
        h1[(gRow0 + r + 8 * laneHi) * HID + col0 + laneLo] = c[r];
}

// ---------------------------------------------------------------------------
// Layer-1 scatter: 16 threads per edge, each moves a float4 of features.
// acc1[col] += h1[row] * norm[e]
// ---------------------------------------------------------------------------
__global__ void gcn_scatter1_kernel(const long long* __restrict__ ei,
                                    const float* __restrict__ norm,
                                    const float* __restrict__ h1,
                                    float* __restrict__ acc1) {
    unsigned gid = blockIdx.x * blockDim.x + threadIdx.x;
    if (gid >= (unsigned)N_EDGES * 16u) return;
    const int e  = (int)(gid >> 4);
    const int f4 = (int)(gid & 15u) * 4;
    const int r  = (int)ei[e];
    const int cn = (int)ei[N_EDGES + e];
    const float s = norm[e];
    const float4 hv = *(const float4*)(h1 + (size_t)r * HID + f4);
    float* dst = acc1 + (size_t)cn * HID + f4;
    atomicAdd(dst + 0, hv.x * s);
    atomicAdd(dst + 1, hv.y * s);
    atomicAdd(dst + 2, hv.z * s);
    atomicAdd(dst + 3, hv.w * s);
}

// Epilogue 1: acc1 = relu(acc1 + b1 + h1 * dinv^2)  (self-loop term folded in)
__global__ void gcn_epi1_kernel(const float* __restrict__ h1,
                                const float* __restrict__ dinv,
                                const float* __restrict__ b1,
                                float* __restrict__ acc1) {
    int gid = blockIdx.x * blockDim.x + threadIdx.x;
    if (gid < N_NODES * HID) {
        const int node = gid >> 6;
        const int f    = gid & 63;
        const float dv = dinv[node];
        float v = acc1[gid] + h1[gid] * dv * dv + b1[f];
        acc1[gid] = v > 0.0f ? v : 0.0f;
    }
}

// GEMM2: t = relu_h @ W2   ([100000,64] @ [64,2]) — tiny, plain dot products.
__global__ void gcn_gemm2_kernel(const float* __restrict__ hrelu,
                                 const float* __restrict__ W2,
                                 float* __restrict__ t) {
    int gid = blockIdx.x * blockDim.x + threadIdx.x;
    if (gid < N_NODES * OUT_C) {
        const int node = gid >> 1;
        const int oc   = gid & 1;
        const float* hr = hrelu + (size_t)node * HID;
        float s = 0.0f;
#pragma unroll
        for (int k = 0; k < HID; ++k) s += hr[k] * W2[k * OUT_C + oc];
        t[gid] = s;
    }
}

// Layer-2 scatter: out[col] += t[row] * norm[e]  (2 floats per edge)
__global__ void gcn_scatter2_kernel(const long long* __restrict__ ei,
                                    const float* __restrict__ norm,
                                    const float* __restrict__ t,
                                    float* __restrict__ out) {
    int e = blockIdx.x * blockDim.x + threadIdx.x;
    if (e < N_EDGES) {
        const int r  = (int)ei[e];
        const int cn = (int)ei[N_EDGES + e];
        const float s = norm[e];
        const float2 tv = *(const float2*)(t + (size_t)r * OUT_C);
        atomicAdd(out + (size_t)cn * OUT_C + 0, tv.x * s);
        atomicAdd(out + (size_t)cn * OUT_C + 1, tv.y * s);
    }
}

// Epilogue 2: out += b2 + t * dinv^2
__global__ void gcn_epi2_kernel(const float* __restrict__ t,
                                const float* __restrict__ dinv,
                                const float* __restrict__ b2,
                                float* __restrict__ out) {
    int gid = blockIdx.x * blockDim.x + threadIdx.x;
    if (gid < N_NODES * OUT_C) {
        const int node = gid >> 1;
        const int f    = gid & 1;
        const float dv = dinv[node];
        out[gid] += t[gid] * dv * dv + b2[f];
    }
}

// ---------------------------------------------------------------------------
extern "C" void kernel_launch(void* const* d_in, const int* in_sizes, int n_in,
                              void* d_out, int out_size, void* d_ws, size_t ws_size,
                              hipStream_t stream) {
    (void)in_sizes; (void)n_in; (void)out_size; (void)ws_size;
    const float*     x  = (const float*)d_in[0];
    const long long* ei = (const long long*)d_in[1];   // [2, E] flattened
    const float*     W1 = (const float*)d_in[2];
    const float*     b1 = (const float*)d_in[3];
    const float*     W2 = (const float*)d_in[4];
    const float*     b2 = (const float*)d_in[5];
    float*           out = (float*)d_out;

    char* ws = (char*)d_ws;
    float* dinv = (float*)ws; ws += sizeof(float) * (size_t)N_NODES;        // deg -> dinv
    float* norm = (float*)ws; ws += sizeof(float) * (size_t)N_EDGES;
    float* h1   = (float*)ws; ws += sizeof(float) * (size_t)N_NODES * HID;
    float* acc1 = (float*)ws; ws += sizeof(float) * (size_t)N_NODES * HID;
    float* t    = (float*)ws;

    // Zero all accumulators every call (deterministic; capture-safe memsets).
    hipMemsetAsync(dinv, 0, sizeof(float) * (size_t)N_NODES, stream);
    hipMemsetAsync(acc1, 0, sizeof(float) * (size_t)N_NODES * HID, stream);
    hipMemsetAsync(out,  0, sizeof(float) * (size_t)N_NODES * OUT_C, stream);

    gcn_deg_kernel <<<(N_EDGES + 255) / 256, 256, 0, stream>>>(ei + N_EDGES, dinv);
    gcn_dinv_kernel<<<(N_NODES + 255) / 256, 256, 0, stream>>>(dinv);
    gcn_norm_kernel<<<(N_EDGES + 255) / 256, 256, 0, stream>>>(ei, dinv, norm);

    gcn_gemm1_wmma<<<N_NODES / 32, 256, 0, stream>>>(x, W1, h1);   // 3125 blocks, exact

    gcn_scatter1_kernel<<<(N_EDGES * 16 + 255) / 256, 256, 0, stream>>>(ei, norm, h1, acc1);
    gcn_epi1_kernel<<<(N_NODES * HID + 255) / 256, 256, 0, stream>>>(h1, dinv, b1, acc1);

    gcn_gemm2_kernel<<<(N_NODES * OUT_C + 255) / 256, 256, 0, stream>>>(acc1, W2, t);
    gcn_scatter2_kernel<<<(N_EDGES + 255) / 256, 256, 0, stream>>>(ei, norm, t, out);
    gcn_epi2_kernel<<<(N_NODES * OUT_C + 255) / 256, 256, 0, stream>>>(t, dinv, b2, out);
}